// GATLayer_33517924778170
// MI455X (gfx1250) — compile-verified
//
#include <hip/hip_runtime.h>
#include <hip/hip_bf16.h>

#define N_ATOMS 20000
#define N_BONDS 200000
#define MAX_DEG 12
#define IN_DIM 256
#define BOND_DIM 128
#define OUT_DIM 512
#define N_HEADS 8
#define HEAD_DIM 64
#define NEG_INF -1e9f

typedef float v2f __attribute__((ext_vector_type(2)));
typedef float v8f __attribute__((ext_vector_type(8)));

__device__ __forceinline__ float wave_sum32(float v) {
#pragma unroll
    for (int off = 16; off > 0; off >>= 1)
        v += __shfl_xor(v, off, 32);
    return v;
}

// C[M x 512] = A[M x K] @ W[K x 512] + bias, fp32 via V_WMMA_F32_16X16X4_F32.
// Block = 256 threads = 8 waves; wave w computes a 32x64 C tile:
// rows [32*blockIdx.x, +32), cols [64*w, +64), as 8 independent 16x16
// accumulators (two M sub-tiles share each B operand -> 8 WMMAs per 8 B loads).
template <int K>
__global__ __launch_bounds__(256)
void wmma_gemm_bias(const float* __restrict__ A, const float* __restrict__ W,
                    const float* __restrict__ bias, float* __restrict__ C) {
    const int lane = threadIdx.x & 31;
    const int wave = threadIdx.x >> 5;
    const int half = lane >> 4;   // 0: lanes 0-15, 1: lanes 16-31
    const int l16  = lane & 15;
    const int m_base = blockIdx.x << 5;
    const int n_base = wave << 6;

    v8f acc00 = {}; v8f acc01 = {}; v8f acc02 = {}; v8f acc03 = {};
    v8f acc10 = {}; v8f acc11 = {}; v8f acc12 = {}; v8f acc13 = {};

    // A 16x4 layout: lanes 0-15 hold (K=k0, k0+1), lanes 16-31 hold (k0+2, k0+3)
    const float* __restrict__ arow0 = A + (size_t)(m_base + l16) * K + 2 * half;
    const float* __restrict__ arow1 = arow0 + (size_t)16 * K;
    // B 4x16 layout (mirrored): VGPR0 = row k0+2*half, VGPR1 = row k0+2*half+1
    const float* __restrict__ wcol = W + (size_t)(2 * half) * OUT_DIM + n_base + l16;

#pragma unroll 2
    for (int k0 = 0; k0 < K; k0 += 4) {
        v2f a0, a1;
        a0.x = arow0[k0];
        a0.y = arow0[k0 + 1];
        a1.x = arow1[k0];
        a1.y = arow1[k0 + 1];
        const float* wp = wcol + (size_t)k0 * OUT_DIM;
        v2f b0, b1, b2, b3;
        b0.x = wp[0];  b0.y = wp[OUT_DIM];
        b1.x = wp[16]; b1.y = wp[OUT_DIM + 16];
        b2.x = wp[32]; b2.y = wp[OUT_DIM + 32];
        b3.x = wp[48]; b3.y = wp[OUT_DIM + 48];
        acc00 = __builtin_amdgcn_wmma_f32_16x16x4_f32(false, a0, false, b0, (short)0, acc00, false, false);
        acc10 = __builtin_amdgcn_wmma_f32_16x16x4_f32(false, a1, false, b0, (short)0, acc10, false, false);
        acc01 = __builtin_amdgcn_wmma_f32_16x16x4_f32(false, a0, false, b1, (short)0, acc01, false, false);
        acc11 = __builtin_amdgcn_wmma_f32_16x16x4_f32(false, a1, false, b1, (short)0, acc11, false, false);
        acc02 = __builtin_amdgcn_wmma_f32_16x16x4_f32(false, a0, false, b2, (short)0, acc02, false, false);
        acc12 = __builtin_amdgcn_wmma_f32_16x16x4_f32(false, a1, false, b2, (short)0, acc12, false, false);
        acc03 = __builtin_amdgcn_wmma_f32_16x16x4_f32(false, a0, false, b3, (short)0, acc03, false, false);
        acc13 = __builtin_amdgcn_wmma_f32_16x16x4_f32(false, a1, false, b3, (short)0, acc13, false, false);
    }

    const float bi0 = bias[n_base + l16];
    const float bi1 = bias[n_base + 16 + l16];
    const float bi2 = bias[n_base + 32 + l16];
    const float bi3 = bias[n_base + 48 + l16];

    // C/D layout: VGPR r -> row r (lanes 0-15) / row 8+r (lanes 16-31)
#pragma unroll
    for (int r = 0; r < 8; ++r) {
        const int row0 = m_base + r + 8 * half;
        float* crow0 = C + (size_t)row0 * OUT_DIM + n_base + l16;
        crow0[0]  = acc00[r] + bi0;
        crow0[16] = acc01[r] + bi1;
        crow0[32] = acc02[r] + bi2;
        crow0[48] = acc03[r] + bi3;
        float* crow1 = crow0 + (size_t)16 * OUT_DIM;
        crow1[0]  = acc10[r] + bi0;
        crow1[16] = acc11[r] + bi1;
        crow1[32] = acc12[r] + bi2;
        crow1[48] = acc13[r] + bi3;
    }
}

// One block per atom (256 thr = 8 wave32), one wave per head, 2 channels/lane.
__global__ __launch_bounds__(256)
void gat_attention(const float* __restrict__ hA, const float* __restrict__ hB,
                   const long long* __restrict__ a2a, const long long* __restrict__ a2b,
                   const float* __restrict__ att_src, const float* __restrict__ att_dst,
                   const float* __restrict__ att_edge, float* __restrict__ out) {
    const int n    = blockIdx.x;
    const int h    = threadIdx.x >> 5;
    const int lane = threadIdx.x & 31;
    const int base = h * HEAD_DIM + lane * 2;   // offset into a 512-wide row

    const v2f asrc = *(const v2f*)(att_src + base);
    const v2f adst = *(const v2f*)(att_dst + base);
    const v2f aedg = *(const v2f*)(att_edge + base);

    const v2f hn = *(const v2f*)(hA + (size_t)n * OUT_DIM + base);
    const float s_src = wave_sum32(hn.x * asrc.x + hn.y * asrc.y);

    float sc[MAX_DEG];
    float vx[MAX_DEG], vy[MAX_DEG];

#pragma unroll
    for (int d = 0; d < MAX_DEG; ++d) {
        const long long j = a2a[(size_t)n * MAX_DEG + d];
        const long long e = a2b[(size_t)n * MAX_DEG + d];
        const v2f ha = *(const v2f*)(hA + (size_t)j * OUT_DIM + base);
        const v2f hb = *(const v2f*)(hB + (size_t)e * OUT_DIM + base);
        vx[d] = ha.x + hb.x;
        vy[d] = ha.y + hb.y;
        const float p = ha.x * adst.x + ha.y * adst.y + hb.x * aedg.x + hb.y * aedg.y;
        float s = s_src + wave_sum32(p);
        s = (s > 0.0f) ? s : 0.2f * s;            // leaky_relu(0.2)
        sc[d] = (j != 0) ? s : NEG_INF;           // atom 0 = padding sentinel
    }

    float m = sc[0];
#pragma unroll
    for (int d = 1; d < MAX_DEG; ++d) m = fmaxf(m, sc[d]);

    float w[MAX_DEG];
    float denom = 0.0f;
#pragma unroll
    for (int d = 0; d < MAX_DEG; ++d) { w[d] = __expf(sc[d] - m); denom += w[d]; }
    const float inv = 1.0f / denom;

    float ox = 0.0f, oy = 0.0f;
#pragma unroll
    for (int d = 0; d < MAX_DEG; ++d) {
        const float al = w[d] * inv;
        ox = fmaf(al, vx[d], ox);
        oy = fmaf(al, vy[d], oy);
    }
    v2f o; o.x = ox; o.y = oy;
    *(v2f*)(out + (size_t)n * OUT_DIM + base) = o;
}

extern "C" void kernel_launch(void* const* d_in, const int* in_sizes, int n_in,
                              void* d_out, int out_size, void* d_ws, size_t ws_size,
                              hipStream_t stream) {
    const float* x         = (const float*)d_in[0];
    const float* edge_attr = (const float*)d_in[1];
    const float* W_atom    = (const float*)d_in[2];
    const float* b_atom    = (const float*)d_in[3];
    const float* W_bond    = (const float*)d_in[4];
    const float* b_bond    = (const float*)d_in[5];
    const float* att_src   = (const float*)d_in[6];
    const float* att_dst   = (const float*)d_in[7];
    const float* att_edge  = (const float*)d_in[8];
    const long long* a2a   = (const long long*)d_in[9];   // int64 in reference
    const long long* a2b   = (const long long*)d_in[10];  // int64 in reference
    float* out = (float*)d_out;

    float* hA = (float*)d_ws;                              // [N_ATOMS, 512]
    float* hB = hA + (size_t)N_ATOMS * OUT_DIM;            // [N_BONDS, 512]

    // 20000/32 = 625, 200000/32 = 6250: exact tiling, no tails.
    wmma_gemm_bias<IN_DIM><<<N_ATOMS / 32, 256, 0, stream>>>(x, W_atom, b_atom, hA);
    wmma_gemm_bias<BOND_DIM><<<N_BONDS / 32, 256, 0, stream>>>(edge_attr, W_bond, b_bond, hB);
    gat_attention<<<N_ATOMS, 256, 0, stream>>>(hA, hB, a2a, a2b,
                                               att_src, att_dst, att_edge, out);
}